// WeightedTripletLoss_78838419685576
// MI455X (gfx1250) — compile-verified
//
#include <hip/hip_runtime.h>
#include <math.h>
#include <stdint.h>

#define TILE 128
#define ROWS_PER_WAVE 16
#define NWAVES 8
#define NTHREADS 256

// Order-preserving float -> uint key so row/col max can use native
// global_atomic_max_u32 (no CAS loop, no FP-atomic flags needed).
__device__ __forceinline__ unsigned f2key(float x) {
    unsigned b = __float_as_uint(x);
    return (b & 0x80000000u) ? ~b : (b | 0x80000000u);
}
__device__ __forceinline__ float key2f(unsigned u) {
    unsigned b = (u & 0x80000000u) ? (u ^ 0x80000000u) : ~u;
    return __uint_as_float(b);
}

__global__ void wtl_init_kernel(unsigned* __restrict__ keys, int n) {
    int i = blockIdx.x * blockDim.x + threadIdx.x;
    if (i < n) keys[i] = 0u;   // below key(-inf); always beaten by real data
}

__global__ __launch_bounds__(NTHREADS)
void wtl_tile_max_kernel(const float* __restrict__ sim,
                         unsigned* __restrict__ rowkey,
                         unsigned* __restrict__ colkey,
                         int B) {
    __shared__ __align__(16) float slab[NWAVES * ROWS_PER_WAVE * TILE]; // 64 KB
    __shared__ __align__(16) float cmred[NWAVES * TILE];                // 4 KB

    const int lane = threadIdx.x & 31;
    const int wave = threadIdx.x >> 5;
    const int col0 = blockIdx.x * TILE;
    const int row0 = blockIdx.y * TILE;
    const int c    = col0 + lane * 4;          // this lane's 4 columns
    const float NEG_INF = -__builtin_inff();

    // ---- Stage this wave's 16 rows (row0+wave, stride 8) into LDS with the
    // ---- CDNA5 async copy engine: one b128 per lane per row segment.
    float* myslab = &slab[wave * ROWS_PER_WAVE * TILE];
    {
        // addrspace(3) pointers flatten as {SHARED_BASE, offset32}: low 32 bits
        // of the generic address are the byte offset within the WG allocation.
        unsigned lds0 = (unsigned)(uintptr_t)myslab + (unsigned)lane * 16u;
        const float* gp = sim + (size_t)(row0 + wave) * (size_t)B
                              + (size_t)col0 + (size_t)lane * 4u;
#pragma unroll
        for (int k = 0; k < ROWS_PER_WAVE; ++k) {
            unsigned long long ga =
                (unsigned long long)(uintptr_t)(gp + (size_t)k * 8u * (size_t)B);
            unsigned la = lds0 + (unsigned)(k * TILE * 4);
            asm volatile("global_load_async_to_lds_b128 %0, %1, off"
                         :: "v"(la), "v"(ga) : "memory");
        }
        asm volatile("s_wait_asynccnt 0" ::: "memory");
    }

    // ---- Consume the slab: per-row wave max (shuffle tree) and per-column
    // ---- register accumulators.
    float4 cm = make_float4(NEG_INF, NEG_INF, NEG_INF, NEG_INF);
#pragma unroll
    for (int k = 0; k < ROWS_PER_WAVE; ++k) {
        const int r = row0 + wave + 8 * k;
        float4 v = *reinterpret_cast<const float4*>(&myslab[k * TILE + lane * 4]);
        // mask the diagonal element (exactly one lane/component in diag tiles)
        if (r == c)     v.x = NEG_INF;
        if (r == c + 1) v.y = NEG_INF;
        if (r == c + 2) v.z = NEG_INF;
        if (r == c + 3) v.w = NEG_INF;

        cm.x = fmaxf(cm.x, v.x);
        cm.y = fmaxf(cm.y, v.y);
        cm.z = fmaxf(cm.z, v.z);
        cm.w = fmaxf(cm.w, v.w);

        float rm = fmaxf(fmaxf(v.x, v.y), fmaxf(v.z, v.w));
#pragma unroll
        for (int off = 16; off > 0; off >>= 1)
            rm = fmaxf(rm, __shfl_xor(rm, off, 32));
        if (lane == 0) atomicMax(&rowkey[r], f2key(rm));
    }

    // ---- Cross-wave column reduction in LDS, then one atomic per column.
    *reinterpret_cast<float4*>(&cmred[wave * TILE + lane * 4]) = cm;
    __syncthreads();
    if (threadIdx.x < TILE) {
        float m = cmred[threadIdx.x];
#pragma unroll
        for (int w = 1; w < NWAVES; ++w)
            m = fmaxf(m, cmred[w * TILE + threadIdx.x]);
        atomicMax(&colkey[col0 + threadIdx.x], f2key(m));
    }
}

__global__ __launch_bounds__(256)
void wtl_finalize_kernel(const float* __restrict__ sim,
                         const unsigned* __restrict__ rowkey,
                         const unsigned* __restrict__ colkey,
                         float* __restrict__ out, int B) {
    __shared__ float red[256];
    float acc = 0.0f;
    for (int i = threadIdx.x; i < B; i += 256) {
        float pos = sim[(size_t)i * (size_t)B + (size_t)i];
        float pl  = fmaxf(0.2f * pos * pos - 0.7f * pos + 0.5f, 0.0f);
        float rm  = key2f(rowkey[i]);
        float cmv = key2f(colkey[i]);
        if (rm + 1.0f > pos)
            acc += pl + fmaxf(0.9f * rm * rm - 0.4f * rm + 0.03f, 0.0f);
        if (cmv + 1.0f > pos)
            acc += pl + fmaxf(0.9f * cmv * cmv - 0.4f * cmv + 0.03f, 0.0f);
    }
    red[threadIdx.x] = acc;
    __syncthreads();
    for (int s = 128; s > 0; s >>= 1) {
        if (threadIdx.x < s) red[threadIdx.x] += red[threadIdx.x + s];
        __syncthreads();
    }
    if (threadIdx.x == 0) out[0] = red[0] / (float)B;
}

extern "C" void kernel_launch(void* const* d_in, const int* in_sizes, int n_in,
                              void* d_out, int out_size, void* d_ws, size_t ws_size,
                              hipStream_t stream) {
    (void)n_in; (void)out_size; (void)ws_size;
    const float* sim = (const float*)d_in[0];
    const int B = (int)(sqrt((double)in_sizes[0]) + 0.5);   // 8192

    unsigned* rowkey = (unsigned*)d_ws;
    unsigned* colkey = rowkey + B;

    wtl_init_kernel<<<(2 * B + 255) / 256, 256, 0, stream>>>(rowkey, 2 * B);

    dim3 grid(B / TILE, B / TILE);   // 64 x 64 = 4096 tiles
    wtl_tile_max_kernel<<<grid, NTHREADS, 0, stream>>>(sim, rowkey, colkey, B);

    wtl_finalize_kernel<<<1, 256, 0, stream>>>(sim, rowkey, colkey, (float*)d_out, B);
}